// PredLayer_75539884802400
// MI455X (gfx1250) — compile-verified
//
#include <hip/hip_runtime.h>

// ---------------------------------------------------------------------------
// PredLayer fused gather/scatter for MI455X (gfx1250, wave32)
//   nc=4, bs=16, h=w=128, oc=48
//   gathered  = einsum('bc,cbhwo->bhwo', softmax(logits*3), states)
//   updated   = states*(1-m) + new*m,  m = softmax(logits*1e10) (~one-hot)
// Single fused streaming pass (~1.0 GB total traffic, HBM-bound).
// Gather einsum runs on V_WMMA_F32_16X16X4_F32.
// ---------------------------------------------------------------------------

typedef float v2f __attribute__((ext_vector_type(2)));
typedef float v8f __attribute__((ext_vector_type(8)));

#define NC 4
#define BS 16
#define NVOL (128 * 128 * 48)      // 786,432 floats per (b)
#define BSN  ((long)BS * NVOL)     // elements per class slab
#define TILES_PER_B (NVOL / 16)    // 49,152 16-float tiles per batch
#define NTILES ((long)BS * TILES_PER_B)

// ---- tiny pre-kernel: per-batch softmax weights into d_ws -----------------
// ws layout: [b][0..3] = gather weights (beta=3), [b][4..7] = scatter masks
__global__ void predlayer_weights_kernel(const float* __restrict__ logits,
                                         float* __restrict__ ws) {
  int b = threadIdx.x;
  if (b >= BS) return;
  float l0 = logits[b * NC + 0];
  float l1 = logits[b * NC + 1];
  float l2 = logits[b * NC + 2];
  float l3 = logits[b * NC + 3];
  float mx = fmaxf(fmaxf(l0, l1), fmaxf(l2, l3));
  // gather weights, beta = 3
  float e0 = __expf((l0 - mx) * 3.0f);
  float e1 = __expf((l1 - mx) * 3.0f);
  float e2 = __expf((l2 - mx) * 3.0f);
  float e3 = __expf((l3 - mx) * 3.0f);
  float inv = 1.0f / (e0 + e1 + e2 + e3);
  ws[b * 8 + 0] = e0 * inv;
  ws[b * 8 + 1] = e1 * inv;
  ws[b * 8 + 2] = e2 * inv;
  ws[b * 8 + 3] = e3 * inv;
  // scatter masks, beta = 1e10 -> exact one-hot (ties split evenly)
  float s0 = expf((l0 - mx) * 1e10f);
  float s1 = expf((l1 - mx) * 1e10f);
  float s2 = expf((l2 - mx) * 1e10f);
  float s3 = expf((l3 - mx) * 1e10f);
  float inv2 = 1.0f / (s0 + s1 + s2 + s3);
  ws[b * 8 + 4] = s0 * inv2;
  ws[b * 8 + 5] = s1 * inv2;
  ws[b * 8 + 6] = s2 * inv2;
  ws[b * 8 + 7] = s3 * inv2;
}

// ---- main fused streaming kernel ------------------------------------------
// One wave processes one 16-float spatial tile per loop step.
//   B-matrix (4x16 f32, 2 VGPRs): v0 lanes0-15=class0, lanes16-31=class1;
//                                 v1 lanes0-15=class2, lanes16-31=class3.
//   A-matrix (16x4 f32, 2 VGPRs): v0 = K0|K2 per half, v1 = K1|K3 -> every
//   row of A equals the gather-weight vector, so every row of D equals G.
//   D VGPR0 holds G[base+n] on both halves -> one full-wave store covers
//   gathered_h (lanes 0-15) and gathered_c (lanes 16-31).
__global__ __launch_bounds__(256) void predlayer_fused_kernel(
    const float* __restrict__ sh, const float* __restrict__ sc,
    const float* __restrict__ nh, const float* __restrict__ ncv,
    const float* __restrict__ ws,
    float* __restrict__ gh, float* __restrict__ gc,
    float* __restrict__ uh, float* __restrict__ uc) {
  const int lane = threadIdx.x & 31;
  const int half = lane >> 4;   // 0: lanes 0-15, 1: lanes 16-31
  const int n    = lane & 15;   // position within 16-float tile

  const long wave   = (long)blockIdx.x * (blockDim.x >> 5) + (threadIdx.x >> 5);
  const long nwaves = (long)gridDim.x * (blockDim.x >> 5);

  for (long t = wave; t < NTILES; t += nwaves) {   // wave-uniform: EXEC stays full
    const int  b    = (int)(t / TILES_PER_B);
    const long base = (t % TILES_PER_B) * 16;
    const long sb   = (long)b * NVOL + base;       // offset within (bs, N)

    const float* wb = ws + b * 8;
    // A-matrix operand: lanes 0-15 carry K=0/1, lanes 16-31 carry K=2/3
    v2f A;
    A.x = wb[half * 2 + 0];
    A.y = wb[half * 2 + 1];

    // B-matrix operands: class 'half' and class '2+half' state tiles (NT loads)
    const long o0 = (long)half * BSN + sb + n;
    const long o1 = (long)(2 + half) * BSN + sb + n;
    float b0h = __builtin_nontemporal_load(sh + o0);
    float b1h = __builtin_nontemporal_load(sh + o1);
    float b0c = __builtin_nontemporal_load(sc + o0);
    float b1c = __builtin_nontemporal_load(sc + o1);

    v2f Bh; Bh.x = b0h; Bh.y = b1h;
    v2f Bc; Bc.x = b0c; Bc.y = b1c;
    v8f Z = {};

    // D[m][n] = sum_k w[k] * s_k[base+n]  (identical for every row m)
    v8f Dh = __builtin_amdgcn_wmma_f32_16x16x4_f32(
        false, A, false, Bh, (short)0, Z, false, false);
    v8f Dc = __builtin_amdgcn_wmma_f32_16x16x4_f32(
        false, A, false, Bc, (short)0, Z, false, false);

    // gather store: lanes 0-15 -> gathered_h, lanes 16-31 -> gathered_c
    {
      float  gv = half ? Dc[0] : Dh[0];
      float* gp = (half ? gc : gh) + sb + n;
      __builtin_nontemporal_store(gv, gp);
    }

    // scatter: reuse in-register state tiles; updated = s + m*(new - s)
    const float m0  = wb[4 + half];        // mask for class 'half'
    const float m1  = wb[4 + 2 + half];    // mask for class '2+half'
    const float nhv = __builtin_nontemporal_load(nh  + sb + n);
    const float ncw = __builtin_nontemporal_load(ncv + sb + n);

    __builtin_nontemporal_store(fmaf(m0, nhv - b0h, b0h), uh + o0);
    __builtin_nontemporal_store(fmaf(m1, nhv - b1h, b1h), uh + o1);
    __builtin_nontemporal_store(fmaf(m0, ncw - b0c, b0c), uc + o0);
    __builtin_nontemporal_store(fmaf(m1, ncw - b1c, b1c), uc + o1);
  }
}

extern "C" void kernel_launch(void* const* d_in, const int* in_sizes, int n_in,
                              void* d_out, int out_size, void* d_ws, size_t ws_size,
                              hipStream_t stream) {
  const float* states_h = (const float*)d_in[0];   // (nc, bs, h, w, oc)
  const float* states_c = (const float*)d_in[1];
  const float* new_h    = (const float*)d_in[2];   // (bs, h, w, oc)
  const float* new_c    = (const float*)d_in[3];
  const float* logits   = (const float*)d_in[4];   // (bs, nc)

  float* out       = (float*)d_out;
  float* gathered_h = out;                          // bs*N
  float* gathered_c = out + BSN;                    // bs*N
  float* updated_h  = out + 2 * BSN;                // nc*bs*N
  float* updated_c  = out + 2 * BSN + NC * BSN;     // nc*bs*N

  float* wbuf = (float*)d_ws;                       // 16*8 floats

  predlayer_weights_kernel<<<1, 32, 0, stream>>>(logits, wbuf);

  // 8192 blocks x 8 waves = 65,536 waves; ~12 tiles each (grid-stride)
  predlayer_fused_kernel<<<8192, 256, 0, stream>>>(
      states_h, states_c, new_h, new_c, wbuf,
      gathered_h, gathered_c, updated_h, updated_c);
}